// HierarchicalPointNet_28097676050525
// MI455X (gfx1250) — compile-verified
//
#include <hip/hip_runtime.h>
#include <math.h>

typedef __attribute__((ext_vector_type(2))) float v2f;
typedef __attribute__((ext_vector_type(8))) float v8f;

#define BN_EPS 1e-5f
#define FPS_THREADS 256

// ---------------- utility ----------------
__global__ void zero_f32(float* __restrict__ p, int n) {
  int i = blockIdx.x * blockDim.x + threadIdx.x;
  if (i < n) p[i] = 0.f;
}

// ---------------- GEMM1: X(rows x 3) @ W(3 x 64) + bias -> Y(rows x 64) ----------------
// One block = 128 threads = 4 waves; block owns 16 rows, wave w owns cols [16w, 16w+16).
// K=3 padded to 4 with zeros. Uses V_WMMA_F32_16X16X4_F32 (fp32 matrix pipe).
__global__ void gemm_k3_n64(const float* __restrict__ X, const float* __restrict__ W,
                            const float* __restrict__ bias, float* __restrict__ Y,
                            int rows) {
  const int lane = threadIdx.x & 31;
  const int wave = threadIdx.x >> 5;      // 0..3
  const int r0   = blockIdx.x * 16;
  const int c0   = wave * 16;
  const int half = lane >> 4;             // 0 or 1
  const int lm   = lane & 15;
  const int kb   = half * 2;              // K base: 0 or 2

  // A: 16x4 (lane<16: K=0,1 ; lane>=16: K=2,3-pad), M = lm
  v2f a;
  a.x = X[(r0 + lm) * 3 + kb];
  a.y = (kb + 1 < 3) ? X[(r0 + lm) * 3 + kb + 1] : 0.f;
  // B: 4x16, N = lm, same K striping
  v2f b;
  b.x = W[kb * 64 + c0 + lm];
  b.y = (kb + 1 < 3) ? W[(kb + 1) * 64 + c0 + lm] : 0.f;

  v8f c = {};
  c = __builtin_amdgcn_wmma_f32_16x16x4_f32(false, a, false, b, (short)0, c, false, false);

  const int col = c0 + lm;
  const float bv = bias[col];
#pragma unroll
  for (int j = 0; j < 8; ++j) {
    int row = r0 + j + half * 8;          // C layout: M = j + 8*half, N = lm
    Y[row * 64 + col] = c[j] + bv;
  }
}

// ---------------- GEMM2: X(rows x 64) @ W(64 x 128) + bias -> Y(rows x 128) ----------------
// One block = 256 threads = 8 waves; block owns 16 rows, wave w owns cols [16w, 16w+16).
// 16 chained k=4 WMMAs accumulate in VGPRs.
__global__ void gemm_k64_n128(const float* __restrict__ X, const float* __restrict__ W,
                              const float* __restrict__ bias, float* __restrict__ Y,
                              int rows) {
  const int lane = threadIdx.x & 31;
  const int wave = threadIdx.x >> 5;      // 0..7
  const int r0   = blockIdx.x * 16;
  const int c0   = wave * 16;
  const int half = lane >> 4;
  const int lm   = lane & 15;

  v8f c = {};
#pragma unroll
  for (int kk = 0; kk < 64; kk += 4) {
    const int k0 = kk + half * 2;
    v2f a, b;
    a.x = X[(r0 + lm) * 64 + k0];
    a.y = X[(r0 + lm) * 64 + k0 + 1];
    b.x = W[k0 * 128 + c0 + lm];
    b.y = W[(k0 + 1) * 128 + c0 + lm];
    c = __builtin_amdgcn_wmma_f32_16x16x4_f32(false, a, false, b, (short)0, c, false, false);
  }
  const int col = c0 + lm;
  const float bv = bias[col];
#pragma unroll
  for (int j = 0; j < 8; ++j)
    Y[(r0 + j + half * 8) * 128 + col] = c[j] + bv;
}

// ---------------- per-channel sum / sum-of-squares (training-mode BN stats) ----------------
// blockDim.x == C; stats[0..C) = sum, stats[C..2C) = sumsq. stats must be pre-zeroed.
__global__ void chan_stats(const float* __restrict__ X, float* __restrict__ stats,
                           int rows, int C) {
  const int c = threadIdx.x;
  float s = 0.f, s2 = 0.f;
  for (int r = blockIdx.x; r < rows; r += gridDim.x) {
    float v = X[(long long)r * C + c];
    s += v; s2 += v * v;
  }
  atomicAdd(&stats[c], s);
  atomicAdd(&stats[C + c], s2);
}

// ---------------- normalize + affine + ReLU, in place ----------------
__global__ void bn_relu(float* __restrict__ X, const float* __restrict__ stats,
                        const float* __restrict__ gamma, const float* __restrict__ beta,
                        int rows, int C) {
  long long i = (long long)blockIdx.x * blockDim.x + threadIdx.x;
  long long total = (long long)rows * C;
  if (i >= total) return;
  int c = (int)(i % C);
  float invN = 1.f / (float)rows;
  float mean = stats[c] * invN;
  float var  = stats[C + c] * invN - mean * mean;   // population variance (ddof=0)
  float v = (X[i] - mean) * rsqrtf(var + BN_EPS) * gamma[c] + beta[c];
  X[i] = v > 0.f ? v : 0.f;
}

// ---------------- global max pool over points; write level slice of output ----------------
__global__ void maxpool(const float* __restrict__ X, float* __restrict__ out,
                        int Npts, int lv) {
  const int b = blockIdx.x;
  const int c = threadIdx.x;              // 128 channels
  const float* p = X + (long long)b * Npts * 128 + c;
  float m = -INFINITY;
  for (int n = 0; n < Npts; ++n) m = fmaxf(m, p[(long long)n * 128]);
  out[b * 384 + lv * 128 + c] = m;
}

// ---------------- farthest point sampling: one block per batch element ----------------
// LDS-resident coords + min-distance array; sequential m-1 iterations of
// (tree-reduction argmax with first-index tie-break) + distance update.
__global__ void fps_kernel(const float* __restrict__ pts, int N, int m,
                           int* __restrict__ idxOut) {
  __shared__ float sx[2048], sy[2048], sz[2048];
  __shared__ float mind[2048];
  __shared__ float rv[FPS_THREADS];
  __shared__ int   ri[FPS_THREADS];
  __shared__ float tgt[3];

  const int b = blockIdx.x;
  const int t = threadIdx.x;
  const float* P = pts + (long long)b * N * 3;

  for (int n = t; n < N; n += FPS_THREADS) {
    sx[n] = P[n * 3 + 0]; sy[n] = P[n * 3 + 1]; sz[n] = P[n * 3 + 2];
  }
  __syncthreads();
  const float x0 = sx[0], y0 = sy[0], z0 = sz[0];
  for (int n = t; n < N; n += FPS_THREADS) {
    float dx = sx[n] - x0, dy = sy[n] - y0, dz = sz[n] - z0;
    mind[n] = dx * dx + dy * dy + dz * dz;
  }
  if (t == 0) idxOut[b * m + 0] = 0;
  __syncthreads();

  for (int i = 1; i < m; ++i) {
    // local argmax over this thread's strided slice (lowest-index tie-break)
    float bv = -1.f; int bi = N;
    for (int n = t; n < N; n += FPS_THREADS) {
      float v = mind[n];
      if (v > bv) { bv = v; bi = n; }
    }
    rv[t] = bv; ri[t] = bi;
    __syncthreads();
    for (int s = FPS_THREADS / 2; s > 0; s >>= 1) {
      if (t < s) {
        float ov = rv[t + s]; int oi = ri[t + s];
        if (ov > rv[t] || (ov == rv[t] && oi < ri[t])) { rv[t] = ov; ri[t] = oi; }
      }
      __syncthreads();
    }
    const int nxt = ri[0];
    if (t == 0) {
      idxOut[b * m + i] = nxt;
      tgt[0] = sx[nxt]; tgt[1] = sy[nxt]; tgt[2] = sz[nxt];
    }
    __syncthreads();
    const float tx = tgt[0], ty = tgt[1], tz = tgt[2];
    for (int n = t; n < N; n += FPS_THREADS) {
      float dx = sx[n] - tx, dy = sy[n] - ty, dz = sz[n] - tz;
      float d = dx * dx + dy * dy + dz * dz;
      mind[n] = fminf(mind[n], d);
    }
    __syncthreads();
  }
}

// ---------------- gather sampled points ----------------
__global__ void gather_pts(const float* __restrict__ pts, const int* __restrict__ idx,
                           float* __restrict__ out, int Nin, int m) {
  const int b = blockIdx.x;
  for (int i = threadIdx.x; i < m; i += blockDim.x) {
    int j = idx[b * m + i];
    const float* src = pts + ((long long)b * Nin + j) * 3;
    float* dst = out + ((long long)b * m + i) * 3;
    dst[0] = src[0]; dst[1] = src[1]; dst[2] = src[2];
  }
}

// ---------------- orchestration ----------------
extern "C" void kernel_launch(void* const* d_in, const int* in_sizes, int n_in,
                              void* d_out, int out_size, void* d_ws, size_t ws_size,
                              hipStream_t stream) {
  const float* points = (const float*)d_in[0];
  const float* w1  = (const float*)d_in[1];
  const float* b1  = (const float*)d_in[2];
  const float* g1  = (const float*)d_in[3];
  const float* be1 = (const float*)d_in[4];
  const float* w2  = (const float*)d_in[5];
  const float* b2  = (const float*)d_in[6];
  const float* g2  = (const float*)d_in[7];
  const float* be2 = (const float*)d_in[8];
  float* out = (float*)d_out;

  const int B = 64, N0 = 2048;

  // workspace layout (floats)
  float* ws    = (float*)d_ws;
  float* h1    = ws;                              // B*N0*64   (33.5 MB)
  float* h2    = h1 + (size_t)B * N0 * 64;        // B*N0*128  (67 MB)
  float* ptsA  = h2 + (size_t)B * N0 * 128;       // B*1024*3
  float* ptsB  = ptsA + (size_t)B * 1024 * 3;     // B*512*3
  float* stats = ptsB + (size_t)B * 512 * 3;      // 256 floats
  int*   fidx  = (int*)(stats + 256);             // B*1024 ints

  const float* P = points;
  int N = N0;
  for (int lv = 0; lv < 3; ++lv) {
    const int rows = B * N;                       // multiple of 16

    // ---- layer 1: 3 -> 64 ----
    zero_f32<<<1, 256, 0, stream>>>(stats, 256);
    gemm_k3_n64<<<rows / 16, 128, 0, stream>>>(P, w1 + lv * 3 * 64, b1 + lv * 64, h1, rows);
    chan_stats<<<512, 64, 0, stream>>>(h1, stats, rows, 64);
    {
      long long tot = (long long)rows * 64;
      bn_relu<<<(unsigned)((tot + 255) / 256), 256, 0, stream>>>(
          h1, stats, g1 + lv * 64, be1 + lv * 64, rows, 64);
    }

    // ---- layer 2: 64 -> 128 ----
    zero_f32<<<1, 256, 0, stream>>>(stats, 256);
    gemm_k64_n128<<<rows / 16, 256, 0, stream>>>(h1, w2 + lv * 64 * 128, b2 + lv * 128, h2, rows);
    chan_stats<<<512, 128, 0, stream>>>(h2, stats, rows, 128);
    {
      long long tot = (long long)rows * 128;
      bn_relu<<<(unsigned)((tot + 255) / 256), 256, 0, stream>>>(
          h2, stats, g2 + lv * 128, be2 + lv * 128, rows, 128);
    }

    // ---- global max pool into output slice ----
    maxpool<<<B, 128, 0, stream>>>(h2, out, N, lv);

    // ---- FPS downsample for next level ----
    if (lv < 2) {
      const int m = N / 2;
      float* nextP = (lv == 0) ? ptsA : ptsB;
      fps_kernel<<<B, FPS_THREADS, 0, stream>>>(P, N, m, fidx);
      gather_pts<<<B, 256, 0, stream>>>(P, fidx, nextP, N, m);
      P = nextP; N = m;
    }
  }
}